// GATEncoder_2757369004087
// MI455X (gfx1250) — compile-verified
//
#include <hip/hip_runtime.h>
#include <hip/hip_bf16.h>
#include <math.h>

#define NEG_SLOPE 0.2f

typedef float v2f __attribute__((ext_vector_type(2)));
typedef float v8f __attribute__((ext_vector_type(8)));

// ---------------------------------------------------------------------------
// C[M,N] = A[M,K] @ W[N,K]^T   (fp32, V_WMMA_F32_16X16X4_F32)
// One wave computes one 16x16 tile. M,N multiples of 16; K multiple of 4.
// A layout (ISA 7.12.2, 32-bit A 16x4): lane L -> row M=L%16, VGPR v -> K=2*(L/16)+v
// B layout (mirrored):                  lane L -> col N=L%16, VGPR v -> K=2*(L/16)+v
// C layout: lane L, VGPR j -> row M=j+8*(L/16), col N=L%16
// ---------------------------------------------------------------------------
__global__ void gemm_wmma_f32(const float* __restrict__ A,
                              const float* __restrict__ W,
                              float* __restrict__ C,
                              int M, int N, int K) {
    const int wavesPerBlock = blockDim.x >> 5;
    const int waveId = blockIdx.x * wavesPerBlock + (threadIdx.x >> 5);
    const int lane   = threadIdx.x & 31;
    const int ntiles = N >> 4;
    const int mtiles = M >> 4;
    if (waveId >= mtiles * ntiles) return;           // wave-uniform
    const int tm = waveId / ntiles;
    const int tn = waveId % ntiles;

    const int rowA  = tm * 16 + (lane & 15);
    const int colB  = tn * 16 + (lane & 15);
    const int khalf = (lane >> 4) * 2;

    v8f acc = {};
    for (int k0 = 0; k0 < K; k0 += 4) {
        v2f a, b;
        a.x = A[(size_t)rowA * K + k0 + khalf + 0];
        a.y = A[(size_t)rowA * K + k0 + khalf + 1];
        b.x = W[(size_t)colB * K + k0 + khalf + 0];
        b.y = W[(size_t)colB * K + k0 + khalf + 1];
        acc = __builtin_amdgcn_wmma_f32_16x16x4_f32(
            /*neg_a=*/false, a, /*neg_b=*/false, b,
            /*c_mod=*/(short)0, acc, /*reuse_a=*/false, /*reuse_b=*/false);
    }

    const int mBase = tm * 16 + 8 * (lane >> 4);
    const int nIdx  = tn * 16 + (lane & 15);
#pragma unroll
    for (int j = 0; j < 8; ++j)
        C[(size_t)(mBase + j) * N + nIdx] = acc[j];
}

// ---------------------------------------------------------------------------
// alpha_src[n,h] = <h[n,h,:], a_src[h,:]> ; same for alpha_dst
// ---------------------------------------------------------------------------
__global__ void node_alpha(const float* __restrict__ h,
                           const float* __restrict__ a_src,
                           const float* __restrict__ a_dst,
                           float* __restrict__ out_s,
                           float* __restrict__ out_d,
                           int N, int H, int Cc) {
    int i = blockIdx.x * blockDim.x + threadIdx.x;
    if (i >= N * H) return;
    int n = i / H, hd = i % H;
    const float* hp = h + ((size_t)n * H + hd) * Cc;
    const float* as = a_src + (size_t)hd * Cc;
    const float* ad = a_dst + (size_t)hd * Cc;
    float ss = 0.f, sd = 0.f;
    for (int c = 0; c < Cc; c += 4) {
        float4 v  = *(const float4*)(hp + c);
        float4 ws = *(const float4*)(as + c);
        float4 wd = *(const float4*)(ad + c);
        ss += v.x * ws.x + v.y * ws.y + v.z * ws.z + v.w * ws.w;
        sd += v.x * wd.x + v.y * wd.y + v.z * wd.z + v.w * wd.w;
    }
    out_s[i] = ss;
    out_d[i] = sd;
}

// ---------------------------------------------------------------------------
// monotone order-preserving float<->uint encoding for atomicMax
// ---------------------------------------------------------------------------
__device__ __forceinline__ unsigned f2ord(float f) {
    unsigned u = __float_as_uint(f);
    return (u & 0x80000000u) ? ~u : (u | 0x80000000u);
}
__device__ __forceinline__ float ord2f(unsigned u) {
    return __uint_as_float((u & 0x80000000u) ? (u ^ 0x80000000u) : ~u);
}

__device__ __forceinline__ void edge_nodes(const int* __restrict__ src,
                                           const int* __restrict__ dst,
                                           int e, int E, int& s, int& d) {
    if (e < E) { s = src[e]; d = dst[e]; }
    else       { s = e - E; d = e - E; }      // appended self-loops
}

__global__ void fill_zero_u32(unsigned* __restrict__ p, size_t n) {
    size_t i = blockIdx.x * (size_t)blockDim.x + threadIdx.x;
    if (i < n) p[i] = 0u;                     // bit pattern 0 == 0.0f too
}

// pass 1: segment max of leaky_relu(a_src[src]+a_dst[dst])
__global__ void edge_max(const int* __restrict__ src, const int* __restrict__ dst,
                         const float* __restrict__ as, const float* __restrict__ ad,
                         unsigned* __restrict__ mbits, int E, int N, int H) {
    int i = blockIdx.x * blockDim.x + threadIdx.x;
    int total = (E + N) * H;
    if (i >= total) return;
    int e = i / H, hd = i % H, s, d;
    edge_nodes(src, dst, e, E, s, d);
    float v = as[(size_t)s * H + hd] + ad[(size_t)d * H + hd];
    v = (v > 0.f) ? v : NEG_SLOPE * v;
    atomicMax(&mbits[(size_t)d * H + hd], f2ord(v));
}

// pass 2: ex = exp(e - m[dst]); denom[dst] += ex; stash ex per (edge,head)
__global__ void edge_exp(const int* __restrict__ src, const int* __restrict__ dst,
                         const float* __restrict__ as, const float* __restrict__ ad,
                         const unsigned* __restrict__ mbits,
                         float* __restrict__ denom, float* __restrict__ alb,
                         int E, int N, int H) {
    int i = blockIdx.x * blockDim.x + threadIdx.x;
    int total = (E + N) * H;
    if (i >= total) return;
    int e = i / H, hd = i % H, s, d;
    edge_nodes(src, dst, e, E, s, d);
    float v = as[(size_t)s * H + hd] + ad[(size_t)d * H + hd];
    v = (v > 0.f) ? v : NEG_SLOPE * v;
    float ex = __expf(v - ord2f(mbits[(size_t)d * H + hd]));
    atomicAdd(&denom[(size_t)d * H + hd], ex);
    alb[i] = ex;
}

// pass 3: alpha = ex / (denom[dst] + eps)
__global__ void alpha_norm(const int* __restrict__ dst,
                           const float* __restrict__ denom,
                           float* __restrict__ alb, int E, int N, int H) {
    int i = blockIdx.x * blockDim.x + threadIdx.x;
    int total = (E + N) * H;
    if (i >= total) return;
    int e = i / H, hd = i % H;
    int d = (e < E) ? dst[e] : (e - E);
    alb[i] = alb[i] / (denom[(size_t)d * H + hd] + 1e-16f);
}

// pass 4: agg[dst,h,:] += alpha * h[src,h,:]   (float4 per thread)
__global__ void edge_aggregate(const int* __restrict__ src, const int* __restrict__ dst,
                               const float* __restrict__ h, const float* __restrict__ alb,
                               float* __restrict__ agg, int E, int N, int H, int Cc) {
    int cg = Cc >> 2;
    long long i = blockIdx.x * (long long)blockDim.x + threadIdx.x;
    long long total = (long long)(E + N) * H * cg;
    if (i >= total) return;
    int c4 = (int)(i % cg);
    long long eh = i / cg;
    int hd = (int)(eh % H);
    int e  = (int)(eh / H);
    int s, d;
    edge_nodes(src, dst, e, E, s, d);
    float a = alb[eh];
    float4 v = *(const float4*)(h + ((size_t)s * H + hd) * Cc + c4 * 4);
    float* o = agg + ((size_t)d * H + hd) * Cc + c4 * 4;
    atomicAdd(o + 0, v.x * a);
    atomicAdd(o + 1, v.y * a);
    atomicAdd(o + 2, v.z * a);
    atomicAdd(o + 3, v.w * a);
}

// x = elu(x + bias)  in place
__global__ void bias_elu(float* __restrict__ x, const float* __restrict__ bias,
                         int N, int F) {
    long long i = blockIdx.x * (long long)blockDim.x + threadIdx.x;
    if (i >= (long long)N * F) return;
    float v = x[i] + bias[i % F];
    x[i] = (v > 0.f) ? v : (__expf(v) - 1.f);
}

// out[n] = <act[n,:64], w> + b
__global__ void final_fc(const float* __restrict__ act, const float* __restrict__ w,
                         const float* __restrict__ b, float* __restrict__ out, int N) {
    int n = blockIdx.x * blockDim.x + threadIdx.x;
    if (n >= N) return;
    float s = 0.f;
#pragma unroll
    for (int c = 0; c < 64; c += 4) {
        float4 v  = *(const float4*)(act + (size_t)n * 64 + c);
        float4 ww = *(const float4*)(w + c);
        s += v.x * ww.x + v.y * ww.y + v.z * ww.z + v.w * ww.w;
    }
    out[n] = s + b[0];
}

// ---------------------------------------------------------------------------
static inline int ceil_div(long long a, int b) { return (int)((a + b - 1) / b); }

extern "C" void kernel_launch(void* const* d_in, const int* in_sizes, int n_in,
                              void* d_out, int out_size, void* d_ws, size_t ws_size,
                              hipStream_t stream) {
    const float* x    = (const float*)d_in[0];
    const int*   ei   = (const int*)d_in[1];     // [2, E] int32
    const float* W1   = (const float*)d_in[2];   // [256,16]
    const float* asr1 = (const float*)d_in[3];   // [4,64]
    const float* ads1 = (const float*)d_in[4];
    const float* b1   = (const float*)d_in[5];   // [256]
    const float* W2   = (const float*)d_in[6];   // [64,256]
    const float* asr2 = (const float*)d_in[7];   // [1,64]
    const float* ads2 = (const float*)d_in[8];
    const float* b2   = (const float*)d_in[9];   // [64]
    const float* fcw  = (const float*)d_in[10];  // [1,64]
    const float* fcb  = (const float*)d_in[11];  // [1]
    float* out = (float*)d_out;

    const int N = in_sizes[0] / 16;              // 50000
    const int E = in_sizes[1] / 2;               // 800000
    const int* src = ei;
    const int* dst = ei + E;

    // ---- workspace layout (fp32 elements) ----
    float* ws = (float*)d_ws;
    size_t off = 0;
    float* h1   = ws + off; off += (size_t)N * 256;
    float* agg1 = ws + off; off += (size_t)N * 256;   // becomes act1 in place
    float* h2   = ws + off; off += (size_t)N * 64;
    float* alb  = ws + off; off += (size_t)(E + N) * 4; // reused for layer 2
    float* as1  = ws + off; off += (size_t)N * 4;
    float* ad1  = ws + off; off += (size_t)N * 4;
    float* den1 = ws + off; off += (size_t)N * 4;
    unsigned* m1 = (unsigned*)(ws + off); off += (size_t)N * 4;
    float* as2  = ws + off; off += (size_t)N;
    float* ad2  = ws + off; off += (size_t)N;
    float* den2 = ws + off; off += (size_t)N;
    unsigned* m2 = (unsigned*)(ws + off); off += (size_t)N;
    float* agg2 = h1;                                  // h1 dead after layer-1 aggregate
    (void)ws_size; (void)n_in; (void)out_size;

    const int B = 256;

    // ================= layer 1 (H=4, C=64) =================
    {
        int tiles = (N / 16) * (256 / 16);            // 50000 waves
        gemm_wmma_f32<<<ceil_div(tiles, 4), 128, 0, stream>>>(x, W1, h1, N, 256, 16);

        node_alpha<<<ceil_div((long long)N * 4, B), B, 0, stream>>>(h1, asr1, ads1, as1, ad1, N, 4, 64);

        fill_zero_u32<<<ceil_div((long long)N * 256, B), B, 0, stream>>>((unsigned*)agg1, (size_t)N * 256);
        fill_zero_u32<<<ceil_div((long long)N * 4, B), B, 0, stream>>>((unsigned*)den1, (size_t)N * 4);
        fill_zero_u32<<<ceil_div((long long)N * 4, B), B, 0, stream>>>(m1, (size_t)N * 4);

        long long eh = (long long)(E + N) * 4;
        edge_max      <<<ceil_div(eh, B), B, 0, stream>>>(src, dst, as1, ad1, m1, E, N, 4);
        edge_exp      <<<ceil_div(eh, B), B, 0, stream>>>(src, dst, as1, ad1, m1, den1, alb, E, N, 4);
        alpha_norm    <<<ceil_div(eh, B), B, 0, stream>>>(dst, den1, alb, E, N, 4);
        edge_aggregate<<<ceil_div(eh * 16, B), B, 0, stream>>>(src, dst, h1, alb, agg1, E, N, 4, 64);

        bias_elu<<<ceil_div((long long)N * 256, B), B, 0, stream>>>(agg1, b1, N, 256);
    }

    // ================= layer 2 (H=1, C=64) =================
    {
        int tiles = (N / 16) * (64 / 16);             // 12500 waves
        gemm_wmma_f32<<<ceil_div(tiles, 4), 128, 0, stream>>>(agg1, W2, h2, N, 64, 256);

        node_alpha<<<ceil_div((long long)N, B), B, 0, stream>>>(h2, asr2, ads2, as2, ad2, N, 1, 64);

        fill_zero_u32<<<ceil_div((long long)N * 64, B), B, 0, stream>>>((unsigned*)agg2, (size_t)N * 64);
        fill_zero_u32<<<ceil_div((long long)N, B), B, 0, stream>>>((unsigned*)den2, (size_t)N);
        fill_zero_u32<<<ceil_div((long long)N, B), B, 0, stream>>>(m2, (size_t)N);

        long long eh = (long long)(E + N);
        edge_max      <<<ceil_div(eh, B), B, 0, stream>>>(src, dst, as2, ad2, m2, E, N, 1);
        edge_exp      <<<ceil_div(eh, B), B, 0, stream>>>(src, dst, as2, ad2, m2, den2, alb, E, N, 1);
        alpha_norm    <<<ceil_div(eh, B), B, 0, stream>>>(dst, den2, alb, E, N, 1);
        edge_aggregate<<<ceil_div(eh * 16, B), B, 0, stream>>>(src, dst, h2, alb, agg2, E, N, 1, 64);

        bias_elu<<<ceil_div((long long)N * 64, B), B, 0, stream>>>(agg2, b2, N, 64);
    }

    final_fc<<<ceil_div((long long)N, B), B, 0, stream>>>(agg2, fcw, fcb, out, N);
}